// mTransformerConv_F_61237643706854
// MI455X (gfx1250) — compile-verified
//
#include <hip/hip_runtime.h>
#include <string.h>

#define N_NODES 100000
#define N_EDGES 1600000
#define NGRAPH  64
#define NHEAD   10

typedef __attribute__((ext_vector_type(2))) float v2f;
typedef __attribute__((ext_vector_type(8))) float v8f;
typedef __attribute__((ext_vector_type(4))) int v4i;

#if defined(__has_builtin)
#if __has_builtin(__builtin_amdgcn_global_load_async_to_lds_b128) && \
    __has_builtin(__builtin_amdgcn_s_wait_asynccnt)
#define HAVE_ASYNC_LDS 1
#endif
#endif

#define AS1 __attribute__((address_space(1)))
#define AS3 __attribute__((address_space(3)))

// ---------- float <-> monotone-unsigned key (for atomic max on floats) ----------
__device__ __forceinline__ unsigned fkey(float f) {
  unsigned b = __float_as_uint(f);
  return (b & 0x80000000u) ? ~b : (b | 0x80000000u);
}
__device__ __forceinline__ float funkey(unsigned k) {
  unsigned b = (k & 0x80000000u) ? (k & 0x7FFFFFFFu) : ~k;
  return __uint_as_float(b);
}

// ---------- cooperative global->LDS copy of 16-byte blocks ----------
__device__ __forceinline__ void stage_to_lds(float* lds_dst, const float* gsrc,
                                             int nfloat4, int tid, int nthreads) {
#ifdef HAVE_ASYNC_LDS
  AS1 v4i* g = (AS1 v4i*)gsrc;
  AS3 v4i* l = (AS3 v4i*)lds_dst;
  for (int i = tid; i < nfloat4; i += nthreads)
    __builtin_amdgcn_global_load_async_to_lds_b128(g + i, l + i, 0, 0);
#else
  const float4* g = (const float4*)gsrc;
  float4* l = (float4*)lds_dst;
  for (int i = tid; i < nfloat4; i += nthreads) l[i] = g[i];
#endif
}
__device__ __forceinline__ void stage_fence() {
#ifdef HAVE_ASYNC_LDS
  __builtin_amdgcn_s_wait_asynccnt(0);
#endif
  __syncthreads();
}

// ---------- generic fill ----------
__global__ void fill_u32(unsigned* __restrict__ p, unsigned v, int n) {
  int i = blockIdx.x * blockDim.x + threadIdx.x;
  if (i < n) p[i] = v;
}

// ---------- zero-pad a weight matrix [rows,cols] -> [prow,cols] ----------
__global__ void pad_w(const float* __restrict__ W, float* __restrict__ Wp,
                      int rows, int cols, int prow) {
  int i = blockIdx.x * blockDim.x + threadIdx.x;
  if (i >= prow * cols) return;
  int r = i / cols;
  Wp[i] = (r < rows) ? W[i] : 0.f;
}

// ============================================================================
// Fused node projection: q1,k1,v1 [N,64-pad] and xr1 [N,5] from x [N,128].
// Block = 256 threads = 8 waves; each wave owns a 16-row M tile.
// B operands (4 padded weight matrices, 128KB) staged in LDS via async copy.
// 13 WMMAs per K-step share one A fragment.
// ============================================================================
__global__ __launch_bounds__(256) void gemm_node_qkvs(
    const float* __restrict__ X,
    const float* __restrict__ Wqp, const float* __restrict__ Wkp,
    const float* __restrict__ Wvp, const float* __restrict__ Wsp,  // [64,128]
    const float* __restrict__ bq, const float* __restrict__ bk,
    const float* __restrict__ bv, const float* __restrict__ bs,
    float* __restrict__ Yq, float* __restrict__ Yk, float* __restrict__ Yv,
    float* __restrict__ XR, int nrows) {
  __shared__ float lw[4 * 64 * 128];  // 128 KB
  int tid = threadIdx.x;
  stage_to_lds(lw + 0 * 8192, Wqp, 2048, tid, 256);
  stage_to_lds(lw + 1 * 8192, Wkp, 2048, tid, 256);
  stage_to_lds(lw + 2 * 8192, Wvp, 2048, tid, 256);
  stage_to_lds(lw + 3 * 8192, Wsp, 2048, tid, 256);
  stage_fence();

  int wave = tid >> 5, lane = tid & 31;
  int row = lane & 15, khalf = lane >> 4;
  int m0 = (blockIdx.x * 8 + wave) * 16;
  int m = m0 + row;
  if (m > nrows - 1) m = nrows - 1;  // branch-free clamp; stores are guarded
  const float* xrow = X + (size_t)m * 128;

  v8f aq[4], ak[4], av[4], as1;
#pragma unroll
  for (int t = 0; t < 4; ++t)
#pragma unroll
    for (int j = 0; j < 8; ++j) { aq[t][j] = 0.f; ak[t][j] = 0.f; av[t][j] = 0.f; }
#pragma unroll
  for (int j = 0; j < 8; ++j) as1[j] = 0.f;

  for (int kk = 0; kk < 32; ++kk) {
    int k0 = kk * 4 + 2 * khalf;
    v2f a;
    a.x = xrow[k0];
    a.y = xrow[k0 + 1];
#pragma unroll
    for (int t = 0; t < 4; ++t) {
      int base = (t * 16 + row) * 128 + k0;
      v2f b;
      b.x = lw[base];         b.y = lw[base + 1];
      aq[t] = __builtin_amdgcn_wmma_f32_16x16x4_f32(false, a, false, b, (short)0, aq[t], false, false);
      b.x = lw[8192 + base];  b.y = lw[8192 + base + 1];
      ak[t] = __builtin_amdgcn_wmma_f32_16x16x4_f32(false, a, false, b, (short)0, ak[t], false, false);
      b.x = lw[16384 + base]; b.y = lw[16384 + base + 1];
      av[t] = __builtin_amdgcn_wmma_f32_16x16x4_f32(false, a, false, b, (short)0, av[t], false, false);
    }
    {
      int base = 24576 + row * 128 + k0;  // skip weight, tile 0 only
      v2f b;
      b.x = lw[base]; b.y = lw[base + 1];
      as1 = __builtin_amdgcn_wmma_f32_16x16x4_f32(false, a, false, b, (short)0, as1, false, false);
    }
  }

#pragma unroll
  for (int t = 0; t < 4; ++t) {
    int col = t * 16 + row;
    float bvq = (col < 50) ? bq[col] : 0.f;
    float bvk = (col < 50) ? bk[col] : 0.f;
    float bvv = (col < 50) ? bv[col] : 0.f;
#pragma unroll
    for (int j = 0; j < 8; ++j) {
      int mr = m0 + j + 8 * khalf;
      if (mr < nrows) {
        Yq[(size_t)mr * 64 + col] = aq[t][j] + bvq;
        Yk[(size_t)mr * 64 + col] = ak[t][j] + bvk;
        Yv[(size_t)mr * 64 + col] = av[t][j] + bvv;
      }
    }
  }
  if (row < 5) {
    float bsv = bs[row];
#pragma unroll
    for (int j = 0; j < 8; ++j) {
      int mr = m0 + j + 8 * khalf;
      if (mr < nrows) XR[(size_t)mr * 5 + row] = as1[j] + bsv;
    }
  }
}

// ============================================================================
// Fused edge projection: e1 [E,64] f16 (4 tiles) + e2 [E,16] f16 (1 tile).
// E = 12500 * 128 exactly; stores unguarded; only a per-wave m0 >= nE exit.
// ============================================================================
__global__ __launch_bounds__(256) void gemm_edge_qe(
    const float* __restrict__ EA, const float* __restrict__ W50p,  // [64,32]
    const float* __restrict__ W10p,                                // [16,32]
    _Float16* __restrict__ O1, _Float16* __restrict__ O2, int nE) {
  __shared__ float lw[(64 + 16) * 32];  // 10 KB
  int tid = threadIdx.x;
  stage_to_lds(lw, W50p, 512, tid, 256);
  stage_to_lds(lw + 2048, W10p, 128, tid, 256);
  stage_fence();

  int wave = tid >> 5, lane = tid & 31;
  int row = lane & 15, khalf = lane >> 4;
  int m0 = (blockIdx.x * 8 + wave) * 16;
  if (m0 >= nE) return;
  const float* arow = EA + (size_t)m0 * 32 + (size_t)row * 32;

  v2f a[8];
#pragma unroll
  for (int kk = 0; kk < 8; ++kk) {
    int k0 = kk * 4 + 2 * khalf;
    a[kk].x = arow[k0];
    a[kk].y = arow[k0 + 1];
  }

  v8f acc[5];
#pragma unroll
  for (int t = 0; t < 5; ++t)
#pragma unroll
    for (int j = 0; j < 8; ++j) acc[t][j] = 0.f;

#pragma unroll
  for (int kk = 0; kk < 8; ++kk) {
    int k0 = kk * 4 + 2 * khalf;
#pragma unroll
    for (int t = 0; t < 4; ++t) {
      int base = (t * 16 + row) * 32 + k0;
      v2f b;
      b.x = lw[base]; b.y = lw[base + 1];
      acc[t] = __builtin_amdgcn_wmma_f32_16x16x4_f32(false, a[kk], false, b, (short)0, acc[t], false, false);
    }
    int base = 2048 + row * 32 + k0;
    v2f b;
    b.x = lw[base]; b.y = lw[base + 1];
    acc[4] = __builtin_amdgcn_wmma_f32_16x16x4_f32(false, a[kk], false, b, (short)0, acc[4], false, false);
  }

#pragma unroll
  for (int t = 0; t < 4; ++t) {
    int col = t * 16 + row;
#pragma unroll
    for (int j = 0; j < 8; ++j) {
      int mr = m0 + j + 8 * khalf;
      O1[(size_t)mr * 64 + col] = (_Float16)acc[t][j];
    }
  }
#pragma unroll
  for (int j = 0; j < 8; ++j) {
    int mr = m0 + j + 8 * khalf;
    O2[(size_t)mr * 16 + row] = (_Float16)acc[4][j];
  }
}

// ---------- conv1 pass A: alpha + segment max (thread per (edge,head)) ----------
__global__ void edge_alpha1(const int* __restrict__ ei, const float* __restrict__ Q,
                            const float* __restrict__ K, const _Float16* __restrict__ E1,
                            float* __restrict__ alpha, unsigned* __restrict__ amax, int nE) {
  unsigned t = blockIdx.x * blockDim.x + threadIdx.x;
  if (t >= (unsigned)nE * NHEAD) return;
  unsigned e = t / NHEAD;
  unsigned h = t - e * NHEAD;
  int s = ei[e], d = ei[(size_t)nE + e];
  const float* q = Q + (size_t)d * 64 + h * 5;
  const float* k = K + (size_t)s * 64 + h * 5;
  const _Float16* ee = E1 + (size_t)e * 64 + h * 5;
  __builtin_prefetch(ee, 0, 1);
  float a = 0.f;
#pragma unroll
  for (int c = 0; c < 5; ++c) a += q[c] * (k[c] + (float)ee[c]);
  a *= 0.44721359549995794f;  // 1/sqrt(5)
  alpha[t] = a;
  atomicMax(&amax[(size_t)d * NHEAD + h], fkey(a));
}

// ---------- generic pass B: ex = exp(alpha - max), denom += ex ----------
__global__ void edge_exp(const int* __restrict__ ei, float* __restrict__ alpha,
                         const unsigned* __restrict__ amax, float* __restrict__ denom, int nE) {
  unsigned t = blockIdx.x * blockDim.x + threadIdx.x;
  if (t >= (unsigned)nE * NHEAD) return;
  unsigned e = t / NHEAD;
  unsigned h = t - e * NHEAD;
  int d = ei[(size_t)nE + e];
  float m = funkey(amax[(size_t)d * NHEAD + h]);
  float ex = __expf(alpha[t] - m);
  alpha[t] = ex;
  atomicAdd(&denom[(size_t)d * NHEAD + h], ex);
}

// ---------- conv1 pass C: out1[dst,c] += mean_h (v[src]+e)*a ----------
__global__ void edge_scatter1(const int* __restrict__ ei, const float* __restrict__ V,
                              const _Float16* __restrict__ E1, const float* __restrict__ alpha,
                              const float* __restrict__ denom, float* __restrict__ out1, int nE) {
  int e = blockIdx.x * blockDim.x + threadIdx.x;
  if (e >= nE) return;
  int s = ei[e], d = ei[(size_t)nE + e];
  __builtin_prefetch(V + (size_t)s * 64, 0, 1);
  float acc[5] = {0.f, 0.f, 0.f, 0.f, 0.f};
#pragma unroll
  for (int h = 0; h < NHEAD; ++h) {
    float a = alpha[(size_t)e * NHEAD + h] / denom[(size_t)d * NHEAD + h];
    const float* vp = V + (size_t)s * 64 + h * 5;
    const _Float16* ep = E1 + (size_t)e * 64 + h * 5;
#pragma unroll
    for (int c = 0; c < 5; ++c) acc[c] += (vp[c] + (float)ep[c]) * a;
  }
#pragma unroll
  for (int c = 0; c < 5; ++c) atomicAdd(&out1[(size_t)d * 5 + c], acc[c] * 0.1f);
}

// ---------- conv1 beta gate + GraphNorm sum ----------
__global__ void node_beta1(const float* __restrict__ out1, const float* __restrict__ xr1,
                           const float* __restrict__ bW, const int* __restrict__ batch,
                           float* __restrict__ h0, float* __restrict__ gnsum,
                           float* __restrict__ gncnt, int n) {
  int i = blockIdx.x * blockDim.x + threadIdx.x;
  if (i >= n) return;
  float o[5], r[5];
#pragma unroll
  for (int c = 0; c < 5; ++c) { o[c] = out1[(size_t)i * 5 + c]; r[c] = xr1[(size_t)i * 5 + c]; }
  float z = 0.f;
#pragma unroll
  for (int c = 0; c < 5; ++c)
    z += bW[c] * o[c] + bW[5 + c] * r[c] + bW[10 + c] * (o[c] - r[c]);
  float beta = 1.f / (1.f + __expf(-z));
  int g = batch[i];
#pragma unroll
  for (int c = 0; c < 5; ++c) {
    float hv = beta * r[c] + (1.f - beta) * o[c];
    h0[(size_t)i * 5 + c] = hv;
    atomicAdd(&gnsum[g * 5 + c], hv);
  }
  atomicAdd(&gncnt[g], 1.f);
}

__global__ void gn_mean(const float* __restrict__ gnsum, const float* __restrict__ gncnt,
                        float* __restrict__ gnmean) {
  int i = threadIdx.x;
  if (i < NGRAPH * 5) gnmean[i] = gnsum[i] / fmaxf(gncnt[i / 5], 1.f);
}

__global__ void node_center(const float* __restrict__ h0, const int* __restrict__ batch,
                            const float* __restrict__ gnmean, const float* __restrict__ ms,
                            float* __restrict__ tmp, float* __restrict__ gnvar, int n) {
  int i = blockIdx.x * blockDim.x + threadIdx.x;
  if (i >= n) return;
  int g = batch[i];
#pragma unroll
  for (int c = 0; c < 5; ++c) {
    float t = h0[(size_t)i * 5 + c] - gnmean[g * 5 + c] * ms[c];
    tmp[(size_t)i * 5 + c] = t;
    atomicAdd(&gnvar[g * 5 + c], t * t);
  }
}

__global__ void gn_rstd(const float* __restrict__ gnvar, const float* __restrict__ gncnt,
                        float* __restrict__ rstd) {
  int i = threadIdx.x;
  if (i < NGRAPH * 5) rstd[i] = rsqrtf(gnvar[i] / fmaxf(gncnt[i / 5], 1.f) + 1e-5f);
}

// ---------- GraphNorm finalize + ReLU + conv2 projections ----------
__global__ void node_norm_proj2(const float* __restrict__ tmp, const int* __restrict__ batch,
                                const float* __restrict__ rstd, const float* __restrict__ gw,
                                const float* __restrict__ gb,
                                const float* __restrict__ q2W, const float* __restrict__ q2b,
                                const float* __restrict__ k2W, const float* __restrict__ k2b,
                                const float* __restrict__ v2W, const float* __restrict__ v2b,
                                const float* __restrict__ s2W, const float* __restrict__ s2b,
                                float* __restrict__ Q2, float* __restrict__ K2,
                                float* __restrict__ V2, float* __restrict__ XR2, int n) {
  int i = blockIdx.x * blockDim.x + threadIdx.x;
  if (i >= n) return;
  int g = batch[i];
  float hv[5];
#pragma unroll
  for (int c = 0; c < 5; ++c)
    hv[c] = fmaxf(0.f, gw[c] * tmp[(size_t)i * 5 + c] * rstd[g * 5 + c] + gb[c]);
  float sk = s2b[0];
#pragma unroll
  for (int c = 0; c < 5; ++c) sk += hv[c] * s2W[c];
  XR2[i] = sk;
#pragma unroll
  for (int h = 0; h < NHEAD; ++h) {
    float aq = q2b[h], ak = k2b[h], av = v2b[h];
#pragma unroll
    for (int c = 0; c < 5; ++c) {
      aq += hv[c] * q2W[h * 5 + c];
      ak += hv[c] * k2W[h * 5 + c];
      av += hv[c] * v2W[h * 5 + c];
    }
    Q2[(size_t)i * NHEAD + h] = aq;
    K2[(size_t)i * NHEAD + h] = ak;
    V2[(size_t)i * NHEAD + h] = av;
  }
}

// ---------- conv2 pass A ----------
__global__ void edge_alpha2(const int* __restrict__ ei, const float* __restrict__ Q2,
                            const float* __restrict__ K2, const _Float16* __restrict__ E2,
                            float* __restrict__ alpha, unsigned* __restrict__ amax, int nE) {
  unsigned t = blockIdx.x * blockDim.x + threadIdx.x;
  if (t >= (unsigned)nE * NHEAD) return;
  unsigned e = t / NHEAD;
  unsigned h = t - e * NHEAD;
  int s = ei[e], d = ei[(size_t)nE + e];
  float a = Q2[(size_t)d * NHEAD + h] *
            (K2[(size_t)s * NHEAD + h] + (float)E2[(size_t)e * 16 + h]);
  alpha[t] = a;
  atomicMax(&amax[(size_t)d * NHEAD + h], fkey(a));
}

// ---------- conv2 pass C ----------
__global__ void edge_scatter2(const int* __restrict__ ei, const float* __restrict__ V2,
                              const _Float16* __restrict__ E2, const float* __restrict__ alpha,
                              const float* __restrict__ denom, float* __restrict__ out2, int nE) {
  int e = blockIdx.x * blockDim.x + threadIdx.x;
  if (e >= nE) return;
  int s = ei[e], d = ei[(size_t)nE + e];
  float acc = 0.f;
#pragma unroll
  for (int h = 0; h < NHEAD; ++h) {
    float a = alpha[(size_t)e * NHEAD + h] / denom[(size_t)d * NHEAD + h];
    acc += (V2[(size_t)s * NHEAD + h] + (float)E2[(size_t)e * 16 + h]) * a;
  }
  atomicAdd(&out2[d], acc * 0.1f);
}

// ---------- final beta gate + sigmoid ----------
__global__ void node_final(const float* __restrict__ out2, const float* __restrict__ xr2,
                           const float* __restrict__ b2W, float* __restrict__ y, int n) {
  int i = blockIdx.x * blockDim.x + threadIdx.x;
  if (i >= n) return;
  float o = out2[i], r = xr2[i];
  float z = b2W[0] * o + b2W[1] * r + b2W[2] * (o - r);
  float beta = 1.f / (1.f + __expf(-z));
  float v = beta * r + (1.f - beta) * o;
  y[i] = 1.f / (1.f + __expf(-v));
}

extern "C" void kernel_launch(void* const* d_in, const int* in_sizes, int n_in,
                              void* d_out, int out_size, void* d_ws, size_t ws_size,
                              hipStream_t stream) {
  const float* x    = (const float*)d_in[0];
  const float* ea   = (const float*)d_in[1];
  const int* ei     = (const int*)d_in[2];
  const int* batch  = (const int*)d_in[3];
  const float* q1W  = (const float*)d_in[4];
  const float* q1b  = (const float*)d_in[5];
  const float* k1W  = (const float*)d_in[6];
  const float* k1b  = (const float*)d_in[7];
  const float* v1W  = (const float*)d_in[8];
  const float* v1b  = (const float*)d_in[9];
  const float* e1W  = (const float*)d_in[10];
  const float* s1W  = (const float*)d_in[11];
  const float* s1b  = (const float*)d_in[12];
  const float* b1W  = (const float*)d_in[13];
  const float* gnW  = (const float*)d_in[14];
  const float* gnb  = (const float*)d_in[15];
  const float* gnms = (const float*)d_in[16];
  const float* q2W  = (const float*)d_in[17];
  const float* q2b  = (const float*)d_in[18];
  const float* k2W  = (const float*)d_in[19];
  const float* k2b  = (const float*)d_in[20];
  const float* v2W  = (const float*)d_in[21];
  const float* v2b  = (const float*)d_in[22];
  const float* e2W  = (const float*)d_in[23];
  const float* s2W  = (const float*)d_in[24];
  const float* s2b  = (const float*)d_in[25];
  const float* b2W  = (const float*)d_in[26];
  float* out        = (float*)d_out;

  size_t off = 0;
  char* base = (char*)d_ws;
  auto alloc = [&](size_t bytes) -> char* {
    char* p = base + off;
    off = (off + bytes + 255) & ~(size_t)255;
    return p;
  };
  float*     q1    = (float*)alloc((size_t)N_NODES * 64 * 4);
  float*     k1    = (float*)alloc((size_t)N_NODES * 64 * 4);
  float*     v1    = (float*)alloc((size_t)N_NODES * 64 * 4);
  _Float16*  e1h   = (_Float16*)alloc((size_t)N_EDGES * 64 * 2);
  _Float16*  e2h   = (_Float16*)alloc((size_t)N_EDGES * 16 * 2);
  float*     alpha = (float*)alloc((size_t)N_EDGES * NHEAD * 4);
  unsigned*  amax  = (unsigned*)alloc((size_t)N_NODES * NHEAD * 4);
  float*     denom = (float*)alloc((size_t)N_NODES * NHEAD * 4);
  float*     xr1   = (float*)alloc((size_t)N_NODES * 5 * 4);
  float*     out1  = (float*)alloc((size_t)N_NODES * 5 * 4);
  float*     h0    = (float*)alloc((size_t)N_NODES * 5 * 4);
  float*     tmpc  = (float*)alloc((size_t)N_NODES * 5 * 4);
  float*     Q2    = (float*)alloc((size_t)N_NODES * NHEAD * 4);
  float*     K2    = (float*)alloc((size_t)N_NODES * NHEAD * 4);
  float*     V2    = (float*)alloc((size_t)N_NODES * NHEAD * 4);
  float*     XR2   = (float*)alloc((size_t)N_NODES * 4);
  float*     out2  = (float*)alloc((size_t)N_NODES * 4);
  float*     gnsum = (float*)alloc(NGRAPH * 5 * 4);   // contiguous with gncnt/gnvar
  float*     gncnt = (float*)alloc(NGRAPH * 4);
  float*     gnvar = (float*)alloc(NGRAPH * 5 * 4);
  float*     gnmean= (float*)alloc(NGRAPH * 5 * 4);
  float*     grstd = (float*)alloc(NGRAPH * 5 * 4);
  float*     Wqp   = (float*)alloc(64 * 128 * 4);
  float*     Wkp   = (float*)alloc(64 * 128 * 4);
  float*     Wvp   = (float*)alloc(64 * 128 * 4);
  float*     Wsp   = (float*)alloc(64 * 128 * 4);
  float*     W50p  = (float*)alloc(64 * 32 * 4);
  float*     W10p  = (float*)alloc(16 * 32 * 4);

  float negbig = -3.0e38f;
  unsigned nb;
  memcpy(&nb, &negbig, 4);
  unsigned keyneg = (nb & 0x80000000u) ? ~nb : (nb | 0x80000000u);

  const int T = 256;
  int nNH = N_NODES * NHEAD;
  unsigned nEH = (unsigned)N_EDGES * NHEAD;
  int gNH = (nNH + T - 1) / T;
  int gEH = (int)((nEH + T - 1) / T);
  int gN  = (N_NODES + T - 1) / T;
  int gE  = (N_EDGES + T - 1) / T;

  // ---- init (conv1) + padded weights ----
  fill_u32<<<gNH, T, 0, stream>>>(amax, keyneg, nNH);
  fill_u32<<<gNH, T, 0, stream>>>((unsigned*)denom, 0u, nNH);
  fill_u32<<<(N_NODES * 5 + T - 1) / T, T, 0, stream>>>((unsigned*)out1, 0u, N_NODES * 5);
  fill_u32<<<3, 256, 0, stream>>>((unsigned*)gnsum, 0u, NGRAPH * 5 + NGRAPH + NGRAPH * 5);
  pad_w<<<32, 256, 0, stream>>>(q1W, Wqp, 50, 128, 64);
  pad_w<<<32, 256, 0, stream>>>(k1W, Wkp, 50, 128, 64);
  pad_w<<<32, 256, 0, stream>>>(v1W, Wvp, 50, 128, 64);
  pad_w<<<32, 256, 0, stream>>>(s1W, Wsp, 5, 128, 64);
  pad_w<<<8, 256, 0, stream>>>(e1W, W50p, 50, 32, 64);
  pad_w<<<2, 256, 0, stream>>>(e2W, W10p, 10, 32, 16);

  // ---- dense projections (WMMA + LDS-staged B) ----
  int gNodeTiles = (N_NODES + 127) / 128;   // 782 blocks of 8 waves
  gemm_node_qkvs<<<gNodeTiles, 256, 0, stream>>>(x, Wqp, Wkp, Wvp, Wsp,
                                                 q1b, k1b, v1b, s1b,
                                                 q1, k1, v1, xr1, N_NODES);
  int gEdgeTiles = N_EDGES / 128;           // 12500, exact
  gemm_edge_qe<<<gEdgeTiles, 256, 0, stream>>>(ea, W50p, W10p, e1h, e2h, N_EDGES);

  // ---- conv1 segment softmax + scatter ----
  edge_alpha1<<<gEH, T, 0, stream>>>(ei, q1, k1, e1h, alpha, amax, N_EDGES);
  edge_exp<<<gEH, T, 0, stream>>>(ei, alpha, amax, denom, N_EDGES);
  edge_scatter1<<<gE, T, 0, stream>>>(ei, v1, e1h, alpha, denom, out1, N_EDGES);

  // ---- conv1 beta + GraphNorm + relu + conv2 projections ----
  node_beta1<<<gN, T, 0, stream>>>(out1, xr1, b1W, batch, h0, gnsum, gncnt, N_NODES);
  gn_mean<<<1, 320, 0, stream>>>(gnsum, gncnt, gnmean);
  node_center<<<gN, T, 0, stream>>>(h0, batch, gnmean, gnms, tmpc, gnvar, N_NODES);
  gn_rstd<<<1, 320, 0, stream>>>(gnvar, gncnt, grstd);
  node_norm_proj2<<<gN, T, 0, stream>>>(tmpc, batch, grstd, gnW, gnb,
                                        q2W, q2b, k2W, k2b, v2W, v2b, s2W, s2b,
                                        Q2, K2, V2, XR2, N_NODES);

  // ---- init (conv2, reuse buffers) ----
  fill_u32<<<gNH, T, 0, stream>>>(amax, keyneg, nNH);
  fill_u32<<<gNH, T, 0, stream>>>((unsigned*)denom, 0u, nNH);
  fill_u32<<<gN, T, 0, stream>>>((unsigned*)out2, 0u, N_NODES);

  // ---- conv2 segment softmax + scatter ----
  edge_alpha2<<<gEH, T, 0, stream>>>(ei, Q2, K2, e2h, alpha, amax, N_EDGES);
  edge_exp<<<gEH, T, 0, stream>>>(ei, alpha, amax, denom, N_EDGES);
  edge_scatter2<<<gE, T, 0, stream>>>(ei, V2, e2h, alpha, denom, out2, N_EDGES);

  // ---- final ----
  node_final<<<gN, T, 0, stream>>>(out2, XR2, b2W, out, N_NODES);
}